// Evaluate_66735201845638
// MI455X (gfx1250) — compile-verified
//
#include <hip/hip_runtime.h>
#include <hip/hip_bf16.h>

// Problem constants (B=1)
namespace {
constexpr int Cc = 32;
constexpr int Hh = 256;
constexpr int Ww = 512;
constexpr int Kk = 10;
constexpr int HW = Hh * Ww;      // 131072
constexpr float TEMP = 10000.0f;
constexpr int BLOCK = 256;
}

// gfx1250 async global->LDS path (ASYNCcnt), guarded so compile never breaks.
#if defined(__HIP_DEVICE_COMPILE__) && defined(__gfx1250__) && \
    __has_builtin(__builtin_amdgcn_global_load_async_to_lds_b32)
#define USE_ASYNC_LDS 1
#else
#define USE_ASYNC_LDS 0
#endif

#define GLOBAL_AS __attribute__((address_space(1)))
#define LDS_AS    __attribute__((address_space(3)))

__global__ __launch_bounds__(BLOCK) void corr_offset_kernel(
    const float* __restrict__ left,   // [C,H,W]
    const float* __restrict__ right,  // [C,H,W]
    const float* __restrict__ offx,   // [K,H,W]
    const float* __restrict__ offy,   // [K,H,W]
    float* __restrict__ out)          // [2,H,W] (ox then oy)
{
    __shared__ float s_ox[Kk][BLOCK];
    __shared__ float s_oy[Kk][BLOCK];

    const int tid = threadIdx.x;
    const int p = blockIdx.x * BLOCK + tid;     // pixel index
    const int h = p >> 9;                       // W = 512
    const int w = p & (Ww - 1);

    // ---- Stage the 20 per-pixel offset values into LDS (read twice later) ----
#if USE_ASYNC_LDS
    #pragma unroll
    for (int k = 0; k < Kk; ++k) {
        __builtin_amdgcn_global_load_async_to_lds_b32(
            (GLOBAL_AS int*)(offx + k * HW + p),
            (LDS_AS int*)&s_ox[k][tid], 0, 0);
        __builtin_amdgcn_global_load_async_to_lds_b32(
            (GLOBAL_AS int*)(offy + k * HW + p),
            (LDS_AS int*)&s_oy[k][tid], 0, 0);
    }
#if __has_builtin(__builtin_amdgcn_s_wait_asynccnt)
    __builtin_amdgcn_s_wait_asynccnt(0);
#else
    asm volatile("s_wait_asynccnt 0" ::: "memory");
#endif
    asm volatile("" ::: "memory");  // keep LDS reads below the wait
#else
    #pragma unroll
    for (int k = 0; k < Kk; ++k) {
        s_ox[k][tid] = offx[k * HW + p];
        s_oy[k][tid] = offy[k * HW + p];
    }
#endif
    // Each thread reads only its own LDS slots -> wave-local wait suffices.

    // ---- Per-candidate corner indices and (validity-folded) weights ----
    int   i00[Kk], i01[Kk], i10[Kk], i11[Kk];
    float w00[Kk], w01[Kk], w10[Kk], w11[Kk];

    const float wf = (float)w, hf = (float)h;
    #pragma unroll
    for (int k = 0; k < Kk; ++k) {
        const float oxv = s_ox[k][tid];
        const float oyv = s_oy[k][tid];
        // faithful to reference coordinate chain
        const float rx = fminf(fmaxf(wf - oxv, 0.0f), (float)(Ww - 1));
        const float ry = fminf(fmaxf(hf - oyv, 0.0f), (float)(Hh - 1));
        const float gx = (rx - (float)(Ww / 2)) * (1.0f / (float)(Ww / 2));
        const float gy = (ry - (float)(Hh / 2)) * (1.0f / (float)(Hh / 2));
        const float ix = ((gx + 1.0f) * (float)Ww - 1.0f) * 0.5f;
        const float iy = ((gy + 1.0f) * (float)Hh - 1.0f) * 0.5f;

        const float x0f = floorf(ix), y0f = floorf(iy);
        const float wx1 = ix - x0f,   wy1 = iy - y0f;
        const float wx0 = 1.0f - wx1, wy0 = 1.0f - wy1;
        const int x0 = (int)x0f, y0 = (int)y0f;
        const int x1 = x0 + 1,   y1 = y0 + 1;

        const float vx0 = (x0 >= 0 && x0 <= Ww - 1) ? 1.0f : 0.0f;
        const float vx1 = (x1 >= 0 && x1 <= Ww - 1) ? 1.0f : 0.0f;
        const float vy0 = (y0 >= 0 && y0 <= Hh - 1) ? 1.0f : 0.0f;
        const float vy1 = (y1 >= 0 && y1 <= Hh - 1) ? 1.0f : 0.0f;

        const int xc0 = min(max(x0, 0), Ww - 1);
        const int xc1 = min(max(x1, 0), Ww - 1);
        const int yc0 = min(max(y0, 0), Hh - 1);
        const int yc1 = min(max(y1, 0), Hh - 1);

        i00[k] = yc0 * Ww + xc0;  w00[k] = wx0 * wy0 * vx0 * vy0;
        i01[k] = yc0 * Ww + xc1;  w01[k] = wx1 * wy0 * vx1 * vy0;
        i10[k] = yc1 * Ww + xc0;  w10[k] = wx0 * wy1 * vx0 * vy1;
        i11[k] = yc1 * Ww + xc1;  w11[k] = wx1 * wy1 * vx1 * vy1;
    }

    float acc[Kk];
    #pragma unroll
    for (int k = 0; k < Kk; ++k) acc[k] = 0.0f;

    // ---- Channel-outer gather/SAD: keeps one channel's gather window
    //      L0-resident while all K candidates x 4 corners reuse it. ----
    #pragma unroll 1
    for (int c = 0; c < Cc; ++c) {
        const float* __restrict__ rc = right + c * HW;
        const float lv = left[c * HW + p];
        #pragma unroll
        for (int k = 0; k < Kk; ++k) {
            const float s = rc[i00[k]] * w00[k] + rc[i01[k]] * w01[k]
                          + rc[i10[k]] * w10[k] + rc[i11[k]] * w11[k];
            acc[k] += fabsf(lv - s);
        }
    }

    // ---- strength = -mean_c(|.|) * T ; acc*(-T/C) rounds identically ----
    #pragma unroll
    for (int k = 0; k < Kk; ++k) acc[k] *= (-TEMP / (float)Cc);

    float m = acc[0];
    #pragma unroll
    for (int k = 1; k < Kk; ++k) m = fmaxf(m, acc[k]);

    float s = 0.0f, sx = 0.0f, sy = 0.0f;
    #pragma unroll
    for (int k = 0; k < Kk; ++k) {
        const float e = expf(acc[k] - m);
        s  += e;
        sx += e * s_ox[k][tid];  // second read of staged offsets (LDS)
        sy += e * s_oy[k][tid];
    }

    const float inv = 1.0f / s;  // s >= 1 (max term contributes e = 1)
    out[p]      = sx * inv;
    out[HW + p] = sy * inv;
}

extern "C" void kernel_launch(void* const* d_in, const int* in_sizes, int n_in,
                              void* d_out, int out_size, void* d_ws, size_t ws_size,
                              hipStream_t stream) {
    (void)in_sizes; (void)n_in; (void)out_size; (void)d_ws; (void)ws_size;
    const float* left  = (const float*)d_in[0];
    const float* right = (const float*)d_in[1];
    const float* offx  = (const float*)d_in[2];
    const float* offy  = (const float*)d_in[3];
    float* out = (float*)d_out;

    dim3 grid(HW / BLOCK);   // 512 blocks x 256 threads = one thread per pixel
    corr_offset_kernel<<<grid, BLOCK, 0, stream>>>(left, right, offx, offy, out);
}